// GNN_Transformer_10857677325090
// MI455X (gfx1250) — compile-verified
//
#include <hip/hip_runtime.h>
#include <hip/hip_bf16.h>

typedef __attribute__((ext_vector_type(2))) float v2f;
typedef __attribute__((ext_vector_type(4))) float v4f;
typedef __attribute__((ext_vector_type(8))) float v8f;
typedef __attribute__((ext_vector_type(4))) int   v4i;

// ---------------------------------------------------------------------------
// Order-preserving int key for float atomic max (handles negatives).
// ---------------------------------------------------------------------------
__device__ __forceinline__ int f2key(float f) {
  int i = __float_as_int(f);
  return (i >= 0) ? i : (i ^ 0x7fffffff);
}
__device__ __forceinline__ float key2f(int i) {
  return __int_as_float((i >= 0) ? i : (i ^ 0x7fffffff));
}

// ---------------------------------------------------------------------------
// C[M,NC] = A[M,K] @ W[K,NC] + bias   (f32 WMMA 16x16x4, one wave per tile)
// Requires M%16==0, NC%16==0, K%4==0. EXEC stays all-ones per wave.
// A-matrix layout (ISA 7.12.2): lane m=lane&15, K pair = 2*(lane>>4)+{0,1}
// B-matrix: n=lane&15, same K pairing. C/D: VGPR r -> rows r (lanes0-15), r+8.
// ---------------------------------------------------------------------------
__global__ __launch_bounds__(256) void gemm_bias_wmma(
    const float* __restrict__ A, const float* __restrict__ W,
    const float* __restrict__ bias, float* __restrict__ C,
    int M, int K, int NC)
{
  const int lane   = threadIdx.x & 31;
  const int wave   = (blockIdx.x * blockDim.x + threadIdx.x) >> 5;
  const int ntiles = NC >> 4;
  const int mtiles = M >> 4;
  const int mt = wave / ntiles;
  const int nt = wave - mt * ntiles;
  if (mt >= mtiles) return;            // wave-uniform exit

  const int  mn   = lane & 15;         // row of A-tile / col of B-tile
  const int  kh   = (lane >> 4) << 1;  // 0 or 2
  const long arow = (long)(mt * 16 + mn) * K;
  const int  ncol = nt * 16 + mn;

  v8f acc = {};
  for (int k = 0; k < K; k += 4) {
    v2f a, b;
    a.x = A[arow + k + kh];
    a.y = A[arow + k + kh + 1];
    b.x = W[(long)(k + kh)     * NC + ncol];
    b.y = W[(long)(k + kh + 1) * NC + ncol];
    acc = __builtin_amdgcn_wmma_f32_16x16x4_f32(
        /*neg_a=*/false, a, /*neg_b=*/false, b,
        /*c_mod=*/(short)0, acc, /*reuse_a=*/false, /*reuse_b=*/false);
  }
  const float bn = bias[ncol];
  const int rowbase = mt * 16 + ((lane >> 4) << 3);
  #pragma unroll
  for (int r = 0; r < 8; ++r)
    C[(long)(rowbase + r) * NC + ncol] = acc[r] + bn;
}

// ---------------------------------------------------------------------------
// Edge pass 1: alpha[e][h] = <q[dst],k[src]>_h * scale ; atomic segment max
// Rows are 4*C floats = C float4 vectors (16B aligned by construction).
// ---------------------------------------------------------------------------
template <int C>
__global__ __launch_bounds__(256) void edge_alpha_max(
    const float* __restrict__ q, const float* __restrict__ k,
    const int* __restrict__ src, const int* __restrict__ dst,
    float* __restrict__ alpha, int* __restrict__ amax,
    int E, float scale)
{
  int e = blockIdx.x * blockDim.x + threadIdx.x;
  if (e >= E) return;
  const int s = src[e], d = dst[e];
  constexpr int V = C;          // float4 vectors per row (4 heads * C / 4)
  constexpr int VH = C / 4;     // float4 vectors per head
  const v4f* qd = (const v4f*)(q + (long)d * (4 * C));
  const v4f* ks = (const v4f*)(k + (long)s * (4 * C));
  v4f qv[V], kv[V];
  #pragma unroll
  for (int i = 0; i < V; ++i) { qv[i] = qd[i]; kv[i] = ks[i]; }
  v4f out;
  #pragma unroll
  for (int h = 0; h < 4; ++h) {
    float a = 0.f;
    #pragma unroll
    for (int i = 0; i < VH; ++i) {
      v4f qq = qv[h * VH + i], kk = kv[h * VH + i];
      a += qq[0] * kk[0] + qq[1] * kk[1] + qq[2] * kk[2] + qq[3] * kk[3];
    }
    a *= scale;
    out[h] = a;
    atomicMax(&amax[d * 4 + h], f2key(a));
  }
  ((v4f*)alpha)[e] = out;
}

// ---------------------------------------------------------------------------
// Edge pass 2: denom[d][h] += exp(alpha - amax[d])
// ---------------------------------------------------------------------------
__global__ __launch_bounds__(256) void edge_denom(
    const float* __restrict__ alpha, const int* __restrict__ dst,
    const int* __restrict__ amax, float* __restrict__ denom, int E)
{
  int e = blockIdx.x * blockDim.x + threadIdx.x;
  if (e >= E) return;
  const int d = dst[e];
  const v4f al = ((const v4f*)alpha)[e];
  const v4i mk = ((const v4i*)amax)[d];
  #pragma unroll
  for (int h = 0; h < 4; ++h)
    atomicAdd(&denom[d * 4 + h], __expf(al[h] - key2f(mk[h])));
}

// ---------------------------------------------------------------------------
// Edge pass 3: agg[d] += v[src] * softmax_weight
// ---------------------------------------------------------------------------
template <int C>
__global__ __launch_bounds__(256) void edge_aggregate(
    const float* __restrict__ alpha, const float* __restrict__ v,
    const int* __restrict__ src, const int* __restrict__ dst,
    const int* __restrict__ amax, const float* __restrict__ denom,
    float* __restrict__ agg, int E)
{
  int e = blockIdx.x * blockDim.x + threadIdx.x;
  if (e >= E) return;
  const int s = src[e], d = dst[e];
  const v4f al = ((const v4f*)alpha)[e];
  const v4i mk = ((const v4i*)amax)[d];
  const v4f dn = ((const v4f*)denom)[d];
  v4f w;
  #pragma unroll
  for (int h = 0; h < 4; ++h)
    w[h] = __expf(al[h] - key2f(mk[h])) / (dn[h] + 1e-16f);

  constexpr int VH = C / 4;     // float4 vectors per head
  const v4f* vs = (const v4f*)(v + (long)s * (4 * C));
  float* ad = agg + (long)d * (4 * C);
  #pragma unroll
  for (int h = 0; h < 4; ++h) {
    #pragma unroll
    for (int i = 0; i < VH; ++i) {
      v4f vv = vs[h * VH + i];
      atomicAdd(&ad[h * C + i * 4 + 0], vv[0] * w[h]);
      atomicAdd(&ad[h * C + i * 4 + 1], vv[1] * w[h]);
      atomicAdd(&ad[h * C + i * 4 + 2], vv[2] * w[h]);
      atomicAdd(&ad[h * C + i * 4 + 3], vv[3] * w[h]);
    }
  }
}

// ---------------------------------------------------------------------------
// h = relu(agg + skip_proj)   (vectorized, n % 4 == 0 by construction)
// ---------------------------------------------------------------------------
__global__ __launch_bounds__(256) void add_relu4(
    const v4f* __restrict__ agg, const v4f* __restrict__ sk,
    v4f* __restrict__ out, int n4)
{
  int i = blockIdx.x * blockDim.x + threadIdx.x;
  if (i >= n4) return;
  v4f t = agg[i] + sk[i];
  v4f r;
  #pragma unroll
  for (int j = 0; j < 4; ++j) r[j] = fmaxf(t[j], 0.f);
  out[i] = r;
}

__global__ __launch_bounds__(256) void fill_f32(float* p, float v, int n) {
  int i = blockIdx.x * blockDim.x + threadIdx.x;
  if (i < n) p[i] = v;
}
__global__ __launch_bounds__(256) void fill_i32(int* p, int v, int n) {
  int i = blockIdx.x * blockDim.x + threadIdx.x;
  if (i < n) p[i] = v;
}

// ---------------------------------------------------------------------------
// Mean pool accumulation (per node) and final FC (per [b, o])
// ---------------------------------------------------------------------------
__global__ __launch_bounds__(256) void pool_accum(
    const float* __restrict__ h, const int* __restrict__ batch,
    float* __restrict__ psum, float* __restrict__ pcnt, int N)
{
  int i = blockIdx.x * blockDim.x + threadIdx.x;
  if (i >= N) return;
  const int b = batch[i];
  const v4f* hr = (const v4f*)(h + (long)i * 16);
  #pragma unroll
  for (int j = 0; j < 4; ++j) {
    v4f hv = hr[j];
    atomicAdd(&psum[b * 16 + j * 4 + 0], hv[0]);
    atomicAdd(&psum[b * 16 + j * 4 + 1], hv[1]);
    atomicAdd(&psum[b * 16 + j * 4 + 2], hv[2]);
    atomicAdd(&psum[b * 16 + j * 4 + 3], hv[3]);
  }
  atomicAdd(&pcnt[b], 1.0f);
}

__global__ __launch_bounds__(256) void pool_fc(
    const float* __restrict__ psum, const float* __restrict__ pcnt,
    const float* __restrict__ Wfc, const float* __restrict__ bfc,
    float* __restrict__ out, int B)
{
  int t = blockIdx.x * blockDim.x + threadIdx.x;
  if (t >= B * 10) return;
  const int b = t / 10, o = t - b * 10;
  const float inv = 1.f / fmaxf(pcnt[b], 1.f);
  float acc = bfc[o];
  #pragma unroll
  for (int c = 0; c < 16; ++c) acc += psum[b * 16 + c] * inv * Wfc[c * 10 + o];
  out[t] = acc;
}

// ---------------------------------------------------------------------------
extern "C" void kernel_launch(void* const* d_in, const int* in_sizes, int n_in,
                              void* d_out, int out_size, void* d_ws, size_t ws_size,
                              hipStream_t stream)
{
  const float* x     = (const float*)d_in[0];
  const int*   ei    = (const int*)d_in[1];
  const int*   batch = (const int*)d_in[2];
  const float *Wq1 = (const float*)d_in[3],  *bq1 = (const float*)d_in[4];
  const float *Wk1 = (const float*)d_in[5],  *bk1 = (const float*)d_in[6];
  const float *Wv1 = (const float*)d_in[7],  *bv1 = (const float*)d_in[8];
  const float *Ws1 = (const float*)d_in[9],  *bs1 = (const float*)d_in[10];
  const float *Wq2 = (const float*)d_in[11], *bq2 = (const float*)d_in[12];
  const float *Wk2 = (const float*)d_in[13], *bk2 = (const float*)d_in[14];
  const float *Wv2 = (const float*)d_in[15], *bv2 = (const float*)d_in[16];
  const float *Ws2 = (const float*)d_in[17], *bs2 = (const float*)d_in[18];
  const float *Wfc = (const float*)d_in[19], *bfc = (const float*)d_in[20];

  const int N = in_sizes[0] / 128;
  const int E = in_sizes[1] / 2;
  const int B = out_size / 10;
  const int* src = ei;
  const int* dst = ei + E;

  // ---- workspace layout (bump allocator, 256B aligned) ----
  char* p = (char*)d_ws;
  auto alloc = [&](size_t bytes) -> char* {
    char* r = p;
    p += (bytes + 255) & ~(size_t)255;
    return r;
  };
  float* q1    = (float*)alloc((size_t)N * 32 * 4);
  float* k1    = (float*)alloc((size_t)N * 32 * 4);
  float* v1    = (float*)alloc((size_t)N * 32 * 4);
  float* s1    = (float*)alloc((size_t)N * 32 * 4);
  float* alpha = (float*)alloc((size_t)E * 4 * 4);
  int*   amax  = (int*)  alloc((size_t)N * 4 * 4);
  float* denom = (float*)alloc((size_t)N * 4 * 4);
  float* agg1  = (float*)alloc((size_t)N * 32 * 4);
  float* h1    = (float*)alloc((size_t)N * 32 * 4);
  float* q2    = (float*)alloc((size_t)N * 16 * 4);
  float* k2    = (float*)alloc((size_t)N * 16 * 4);
  float* v2    = (float*)alloc((size_t)N * 16 * 4);
  float* s2    = (float*)alloc((size_t)N * 16 * 4);
  float* agg2  = (float*)alloc((size_t)N * 16 * 4);
  float* h2    = (float*)alloc((size_t)N * 16 * 4);
  float* psum  = (float*)alloc((size_t)B * 16 * 4);
  float* pcnt  = (float*)alloc((size_t)B * 4);

  const int TB = 256;
  auto blk  = [](int n) { return (n + 255) / 256; };
  auto gblk = [](int tiles) { return (tiles * 32 + 255) / 256; };  // 8 waves/block

  const float sc1 = 0.35355339059327373f;  // 1/sqrt(8)
  const float sc2 = 0.5f;                  // 1/sqrt(4)
  const int NEG_INF_KEY = (int)0x80000000; // below key(-inf)

  // ================= conv1 =================
  {
    const int tiles = (N / 16) * (32 / 16);
    gemm_bias_wmma<<<gblk(tiles), TB, 0, stream>>>(x, Wq1, bq1, q1, N, 128, 32);
    gemm_bias_wmma<<<gblk(tiles), TB, 0, stream>>>(x, Wk1, bk1, k1, N, 128, 32);
    gemm_bias_wmma<<<gblk(tiles), TB, 0, stream>>>(x, Wv1, bv1, v1, N, 128, 32);
    gemm_bias_wmma<<<gblk(tiles), TB, 0, stream>>>(x, Ws1, bs1, s1, N, 128, 32);

    fill_i32<<<blk(N * 4),  TB, 0, stream>>>(amax,  NEG_INF_KEY, N * 4);
    fill_f32<<<blk(N * 4),  TB, 0, stream>>>(denom, 0.f,         N * 4);
    fill_f32<<<blk(N * 32), TB, 0, stream>>>(agg1,  0.f,         N * 32);

    edge_alpha_max<8><<<blk(E), TB, 0, stream>>>(q1, k1, src, dst, alpha, amax, E, sc1);
    edge_denom      <<<blk(E), TB, 0, stream>>>(alpha, dst, amax, denom, E);
    edge_aggregate<8><<<blk(E), TB, 0, stream>>>(alpha, v1, src, dst, amax, denom, agg1, E);

    add_relu4<<<blk(N * 8), TB, 0, stream>>>((const v4f*)agg1, (const v4f*)s1,
                                             (v4f*)h1, N * 8);
  }

  // ================= conv2 =================
  {
    const int tiles = (N / 16) * (16 / 16);
    gemm_bias_wmma<<<gblk(tiles), TB, 0, stream>>>(h1, Wq2, bq2, q2, N, 32, 16);
    gemm_bias_wmma<<<gblk(tiles), TB, 0, stream>>>(h1, Wk2, bk2, k2, N, 32, 16);
    gemm_bias_wmma<<<gblk(tiles), TB, 0, stream>>>(h1, Wv2, bv2, v2, N, 32, 16);
    gemm_bias_wmma<<<gblk(tiles), TB, 0, stream>>>(h1, Ws2, bs2, s2, N, 32, 16);

    fill_i32<<<blk(N * 4),  TB, 0, stream>>>(amax,  NEG_INF_KEY, N * 4);
    fill_f32<<<blk(N * 4),  TB, 0, stream>>>(denom, 0.f,         N * 4);
    fill_f32<<<blk(N * 16), TB, 0, stream>>>(agg2,  0.f,         N * 16);

    edge_alpha_max<4><<<blk(E), TB, 0, stream>>>(q2, k2, src, dst, alpha, amax, E, sc2);
    edge_denom      <<<blk(E), TB, 0, stream>>>(alpha, dst, amax, denom, E);
    edge_aggregate<4><<<blk(E), TB, 0, stream>>>(alpha, v2, src, dst, amax, denom, agg2, E);

    add_relu4<<<blk(N * 4), TB, 0, stream>>>((const v4f*)agg2, (const v4f*)s2,
                                             (v4f*)h2, N * 4);
  }

  // ================= pool + fc =================
  fill_f32<<<blk(B * 16), TB, 0, stream>>>(psum, 0.f, B * 16);
  fill_f32<<<blk(B),      TB, 0, stream>>>(pcnt, 0.f, B);
  pool_accum<<<blk(N), TB, 0, stream>>>(h2, batch, psum, pcnt, N);
  pool_fc<<<blk(B * 10), TB, 0, stream>>>(psum, pcnt, Wfc, bfc, (float*)d_out, B);
}